// GNN_20117626814606
// MI455X (gfx1250) — compile-verified
//
#include <hip/hip_runtime.h>
#include <cstdint>

#define N_NODES 100000
#define N_EDGES 1600000
#define IN_CH   256
#define HID     128
#define K_STEPS 10
#define ALPHA   0.1f

typedef __attribute__((ext_vector_type(2))) float v2f;
typedef __attribute__((ext_vector_type(8))) float v8f;

__device__ __forceinline__ void atom_add_f32(float* p, float v) {
    // relaxed + agent scope -> non-returning global_atomic_add_f32
    __hip_atomic_fetch_add(p, v, __ATOMIC_RELAXED, __HIP_MEMORY_SCOPE_AGENT);
}

// ---- degree accumulation: one thread per edge ------------------------------
__global__ __launch_bounds__(256) void deg_kernel(const int* __restrict__ src,
                                                  const int* __restrict__ dst,
                                                  float* __restrict__ deg_out,
                                                  float* __restrict__ deg_in) {
    int e = blockIdx.x * blockDim.x + threadIdx.x;
    if (e < N_EDGES) {
        atom_add_f32(&deg_out[src[e]], 1.0f);
        atom_add_f32(&deg_in[dst[e]], 1.0f);
    }
}

// ---- deg -> D^{-1/2} in place (covers both deg arrays in one launch) -------
__global__ __launch_bounds__(256) void inv_kernel(float* __restrict__ deg, int n) {
    int i = blockIdx.x * blockDim.x + threadIdx.x;
    if (i < n) {
        float d = deg[i];
        deg[i] = (d > 0.0f) ? rsqrtf(fmaxf(d, 1.0f)) : 0.0f;
    }
}

// ---- per-edge symmetric norm ----------------------------------------------
__global__ __launch_bounds__(256) void enorm_kernel(const int* __restrict__ src,
                                                    const int* __restrict__ dst,
                                                    const float* __restrict__ inv_out,
                                                    const float* __restrict__ inv_in,
                                                    float* __restrict__ enorm) {
    int e = blockIdx.x * blockDim.x + threadIdx.x;
    if (e < N_EDGES)
        enorm[e] = inv_out[src[e]] * inv_in[dst[e]];
}

// ---- dense GEMM  H = X(100000x256) @ W(256x128) via V_WMMA_F32_16X16X4_F32 -
// Block = 256 threads = 8 waves; blockIdx.x = 16-row tile; wave = 16-col tile.
__global__ __launch_bounds__(256) void gemm_kernel(const float* __restrict__ X,
                                                   const float* __restrict__ W,
                                                   float* __restrict__ H) {
    const int lane    = threadIdx.x & 31;
    const int wave    = threadIdx.x >> 5;
    const int rowBase = blockIdx.x * 16;
    const int colBase = wave * 16;
    const int half    = lane >> 4;   // 0: K pair {0,1}; 1: K pair {2,3}
    const int l16     = lane & 15;

    const float* __restrict__ arow = X + (size_t)(rowBase + l16) * IN_CH + half * 2;
    const float* __restrict__ wcol = W + (size_t)(half * 2) * HID + (colBase + l16);

    v8f acc = {};
#pragma unroll 8
    for (int k0 = 0; k0 < IN_CH; k0 += 4) {
        v2f a, b;
        a.x = arow[k0];
        a.y = arow[k0 + 1];
        b.x = wcol[k0 * HID];
        b.y = wcol[k0 * HID + HID];
        acc = __builtin_amdgcn_wmma_f32_16x16x4_f32(
            /*neg_a=*/false, a, /*neg_b=*/false, b,
            /*c_mod=*/(short)0, acc, /*reuse_a=*/false, /*reuse_b=*/false);
    }

    // C/D layout: VGPR i -> M = i (lanes 0-15) or i+8 (lanes 16-31), N = lane&15
    float* __restrict__ hrow = H + (size_t)(rowBase + half * 8) * HID + (colBase + l16);
#pragma unroll
    for (int i = 0; i < 8; ++i)
        hrow[(size_t)i * HID] = acc[i];
}

// ---- propagation: one wave per edge, lane handles 4 of 128 channels --------
__global__ __launch_bounds__(256) void prop_kernel(const float* __restrict__ h,
                                                   const int* __restrict__ src,
                                                   const int* __restrict__ dst,
                                                   const float* __restrict__ enorm,
                                                   float* __restrict__ out) {
    const int lane = threadIdx.x & 31;
    const int e    = blockIdx.x * 8 + (threadIdx.x >> 5);
    if (e >= N_EDGES) return;
    const int   s = src[e];      // wave-uniform -> scalar loads
    const int   d = dst[e];
    const float w = enorm[e];
    const float4 v = *(const float4*)(h + (size_t)s * HID + lane * 4);
    float* o = out + (size_t)d * HID + lane * 4;
    atom_add_f32(o + 0, v.x * w);
    atom_add_f32(o + 1, v.y * w);
    atom_add_f32(o + 2, v.z * w);
    atom_add_f32(o + 3, v.w * w);
}

// ---- h = relu(h + b), float4 per thread ------------------------------------
__global__ __launch_bounds__(256) void bias_relu_kernel(float* __restrict__ h,
                                                        const float* __restrict__ bias) {
    size_t i = ((size_t)blockIdx.x * blockDim.x + threadIdx.x) * 4;
    if (i >= (size_t)N_NODES * HID) return;
    int col = (int)(i % HID);
    float4 v  = *(float4*)(h + i);
    float4 bb = *(const float4*)(bias + col);
    v.x = fmaxf(v.x + bb.x, 0.0f);
    v.y = fmaxf(v.y + bb.y, 0.0f);
    v.z = fmaxf(v.z + bb.z, 0.0f);
    v.w = fmaxf(v.w + bb.w, 0.0f);
    *(float4*)(h + i) = v;
}

// ---- out = (1-alpha)*p + alpha*h0, float4 per thread -----------------------
__global__ __launch_bounds__(256) void axpy_kernel(const float* __restrict__ p,
                                                   const float* __restrict__ h0,
                                                   float* __restrict__ out) {
    size_t i = ((size_t)blockIdx.x * blockDim.x + threadIdx.x) * 4;
    if (i >= (size_t)N_NODES * HID) return;
    float4 a = *(const float4*)(p + i);
    float4 b = *(const float4*)(h0 + i);
    float4 r;
    r.x = (1.0f - ALPHA) * a.x + ALPHA * b.x;
    r.y = (1.0f - ALPHA) * a.y + ALPHA * b.y;
    r.z = (1.0f - ALPHA) * a.z + ALPHA * b.z;
    r.w = (1.0f - ALPHA) * a.w + ALPHA * b.w;
    *(float4*)(out + i) = r;
}

extern "C" void kernel_launch(void* const* d_in, const int* in_sizes, int n_in,
                              void* d_out, int out_size, void* d_ws, size_t ws_size,
                              hipStream_t stream) {
    const float* features = (const float*)d_in[0];   // [N, 256]
    const float* W        = (const float*)d_in[1];   // [256, 128]
    const float* bias     = (const float*)d_in[2];   // [128]
    const int*   eidx     = (const int*)d_in[3];     // [2, E] row-major
    const int*   src      = eidx;
    const int*   dst      = eidx + N_EDGES;

    float* out = (float*)d_out;                      // [N, 128]

    const size_t NH = (size_t)N_NODES * HID;
    float* deg     = (float*)d_ws;                   // 2N (deg_out | deg_in)
    float* deg_out = deg;
    float* deg_in  = deg + N_NODES;
    float* enorm   = deg + 2 * N_NODES;              // E
    float* bufA    = enorm + N_EDGES;                // N*HID
    float* bufB    = bufA + NH;                      // N*HID  (becomes h0)

    // ---- edge norms ----
    hipMemsetAsync(deg, 0, 2 * (size_t)N_NODES * sizeof(float), stream);
    deg_kernel<<<(N_EDGES + 255) / 256, 256, 0, stream>>>(src, dst, deg_out, deg_in);
    inv_kernel<<<(2 * N_NODES + 255) / 256, 256, 0, stream>>>(deg, 2 * N_NODES);
    enorm_kernel<<<(N_EDGES + 255) / 256, 256, 0, stream>>>(src, dst, deg_out, deg_in, enorm);

    // ---- GraphConv: project (WMMA f32), propagate, bias+relu ----
    gemm_kernel<<<N_NODES / 16, 256, 0, stream>>>(features, W, bufA);
    hipMemsetAsync(bufB, 0, NH * sizeof(float), stream);
    prop_kernel<<<N_EDGES / 8, 256, 0, stream>>>(bufA, src, dst, enorm, bufB);
    bias_relu_kernel<<<(int)(NH / 4 / 256), 256, 0, stream>>>(bufB, bias);

    // ---- APPNP: h0 = bufB; temps alternate bufA / d_out; last lands in d_out
    float* cur = bufB;
    for (int t = 0; t < K_STEPS; ++t) {
        float* tmp = (t & 1) ? out : bufA;
        hipMemsetAsync(tmp, 0, NH * sizeof(float), stream);
        prop_kernel<<<N_EDGES / 8, 256, 0, stream>>>(cur, src, dst, enorm, tmp);
        axpy_kernel<<<(int)(NH / 4 / 256), 256, 0, stream>>>(tmp, bufB, tmp);
        cur = tmp;
    }
    // K_STEPS is even-numbered from 0..9 -> last tmp == out; result already in d_out.
}